// LlamaDecodeAttention_77670188580974
// MI455X (gfx1250) — compile-verified
//
#include <hip/hip_runtime.h>
#include <math.h>

typedef __attribute__((ext_vector_type(2))) float v2f;
typedef __attribute__((ext_vector_type(4))) float v4f;
typedef __attribute__((ext_vector_type(8))) float v8f;

#define DIM 4096
#define NHEADS 32
#define HEADDIM 128
#define CACHELEN 4096
#define BATCH 16
#define NCHUNK 8
#define CHUNK (CACHELEN / NCHUNK)   // 512
#define KSPLIT 8                    // 8 K-segments per 16-col tile
#define SCALE 0.08838834764831845f  // 1/sqrt(128)

// ---------------------------------------------------------------------------
// Zero-fill both atomic-add target regions in ONE launch (q/k_new/v_new in ws
// and d_out), avoiding a serialization bubble between two tiny kernels.
// ---------------------------------------------------------------------------
__global__ void zero_targets(float* __restrict__ ws, float* __restrict__ out) {
  int i = blockIdx.x * blockDim.x + threadIdx.x;
  if (i < 196608) ws[i] = 0.0f;
  else if (i < 196608 + 65536) out[i - 196608] = 0.0f;
}

// ---------------------------------------------------------------------------
// Skinny GEMM core: out(16 x 4096) += A(16 x 4096) @ W(4096 x 4096)^T restricted
// to one 16-column tile and one K-segment.  V_WMMA_F32_16X16X4_F32.
// Lane layout (32-bit A 16x4): lanes 0-15 hold slots K0,K1; lanes 16-31 K2,K3.
// We map slots of wmma#0 -> {k,k+1,k+4,k+5}, wmma#1 -> {k+2,k+3,k+6,k+7} so one
// float4 load per lane feeds two WMMAs (valid: K-sum is commutative, A and B
// use the same slot->k map per instruction).
// ---------------------------------------------------------------------------
__device__ __forceinline__ void gemm16_core(const float* __restrict__ A,
                                            const float* __restrict__ W,
                                            float* __restrict__ out,
                                            int n0, int ks, int lane) {
  const int l16  = lane & 15;
  const int half = lane >> 4;
  const int KSEG = DIM / KSPLIT;        // 512
  const int k0   = ks * KSEG;

  const float* aPtr = A + (size_t)l16 * DIM + k0 + half * 4;
  const float* wPtr = W + (size_t)(n0 + l16) * DIM + k0 + half * 4;

  v8f c = {};
  #pragma unroll 4
  for (int k = 0; k < KSEG; k += 8) {
    v4f a = *(const v4f*)(aPtr + k);                                  // x/att: hot, cached
    v4f b = __builtin_nontemporal_load((const v4f*)(wPtr + k));       // weights: single use
    v2f a0 = {a.x, a.y}, a1 = {a.z, a.w};
    v2f b0 = {b.x, b.y}, b1 = {b.z, b.w};
    c = __builtin_amdgcn_wmma_f32_16x16x4_f32(false, a0, false, b0, (short)0, c, false, false);
    c = __builtin_amdgcn_wmma_f32_16x16x4_f32(false, a1, false, b1, (short)0, c, false, false);
  }

  // C/D layout: VGPR r -> M = r + 8*half, N = l16
  float* o = out + (size_t)(half * 8) * DIM + n0 + l16;
  #pragma unroll
  for (int r = 0; r < 8; ++r)
    unsafeAtomicAdd(o + (size_t)r * DIM, c[r]);   // global_atomic_add_f32 (K-split reduce)
}

// Fused Q/K/V projection: 3 matrices * 256 tiles * KSPLIT waves = 6144 waves
// in ONE launch (~24 MB of loads in flight -> saturates HBM on the 192 MB read).
// grid = 768 blocks x 256 threads.
__global__ void qkv_gemm_fused(const float* __restrict__ x,
                               const float* __restrict__ wq,
                               const float* __restrict__ wk,
                               const float* __restrict__ wv,
                               float* __restrict__ q,
                               float* __restrict__ kn,
                               float* __restrict__ vn) {
  const int lane  = threadIdx.x & 31;
  const int gwave = (blockIdx.x * blockDim.x + threadIdx.x) >> 5;
  const int tile3 = gwave / KSPLIT;     // 0..767
  const int ks    = gwave % KSPLIT;
  const int mat   = tile3 >> 8;         // 0:q 1:k 2:v
  const int n0    = (tile3 & 255) * 16;
  const float* W = (mat == 0) ? wq : (mat == 1) ? wk : wv;
  float*     out = (mat == 0) ? q  : (mat == 1) ? kn : vn;
  gemm16_core(x, W, out, n0, ks, lane);
}

// Single-matrix projection (WO): 256 tiles * KSPLIT = 2048 waves, 256 blocks.
__global__ void gemm16_wmma(const float* __restrict__ A,
                            const float* __restrict__ W,
                            float* __restrict__ out) {
  const int lane  = threadIdx.x & 31;
  const int gwave = (blockIdx.x * blockDim.x + threadIdx.x) >> 5;
  gemm16_core(A, W, out, (gwave / KSPLIT) * 16, gwave % KSPLIT, lane);
}

// ---------------------------------------------------------------------------
// Wave reductions (wave32)
// ---------------------------------------------------------------------------
__device__ __forceinline__ float waveSum(float v) {
  #pragma unroll
  for (int o = 16; o > 0; o >>= 1) v += __shfl_xor(v, o, 32);
  return v;
}
__device__ __forceinline__ float waveMax(float v) {
  #pragma unroll
  for (int o = 16; o > 0; o >>= 1) v = fmaxf(v, __shfl_xor(v, o, 32));
  return v;
}

// ---------------------------------------------------------------------------
// Flash-decode partial: one block per (b,h,chunk). 8 waves x 64 rows.
// Each row: 32 lanes load float4 of K (512B coalesced, NT), dot with q, wave-
// reduce -> logit. Then chunk-local softmax (m, l) and p.V accumulation.
// Outputs: acc_part[bh][chunk][128], ml_part[bh][chunk][2] = {m, l}
// ---------------------------------------------------------------------------
__global__ void attn_partial(const float* __restrict__ q,
                             const float* __restrict__ kc,
                             const float* __restrict__ vc,
                             float* __restrict__ acc_part,
                             float* __restrict__ ml_part) {
  const int bh    = blockIdx.x >> 3;   // 0..511
  const int chunk = blockIdx.x & 7;
  const int b = bh >> 5, h = bh & 31;
  const int lane = threadIdx.x & 31;
  const int wave = threadIdx.x >> 5;

  const size_t base = ((size_t)bh * CACHELEN + (size_t)chunk * CHUNK) * HEADDIM;
  const float* K = kc + base;
  const float* V = vc + base;
  const v4f q4 = *(const v4f*)(q + (size_t)b * DIM + h * HEADDIM + lane * 4);

  __shared__ float logits[CHUNK];
  __shared__ float wred[8];
  __shared__ float accbuf[HEADDIM];
  __shared__ float bcast[2];

  if (threadIdx.x < HEADDIM) accbuf[threadIdx.x] = 0.0f;

  // ---- scores: wave w handles rows [w*64, w*64+64); 4 NT loads in flight ----
  #pragma unroll 4
  for (int i = 0; i < CHUNK / 8; ++i) {
    const int r = wave * (CHUNK / 8) + i;
    v4f k4 = __builtin_nontemporal_load((const v4f*)(K + (size_t)r * HEADDIM + lane * 4));
    float p = q4.x * k4.x + q4.y * k4.y + q4.z * k4.z + q4.w * k4.w;
    p = waveSum(p);
    if (lane == 0) logits[r] = p * SCALE;
  }
  __syncthreads();

  // ---- chunk max ----
  float m = -3.4e38f;
  for (int r = threadIdx.x; r < CHUNK; r += 256) m = fmaxf(m, logits[r]);
  m = waveMax(m);
  if (lane == 0) wred[wave] = m;
  __syncthreads();
  m = wred[0];
  #pragma unroll
  for (int w = 1; w < 8; ++w) m = fmaxf(m, wred[w]);
  __syncthreads();

  // ---- p = exp(logit - m), l = sum p ----
  float lsum = 0.0f;
  for (int r = threadIdx.x; r < CHUNK; r += 256) {
    float p = __expf(logits[r] - m);
    logits[r] = p;
    lsum += p;
  }
  lsum = waveSum(lsum);
  __syncthreads();               // wred reuse + logits overwrite visible
  if (lane == 0) wred[wave] = lsum;
  __syncthreads();
  if (threadIdx.x == 0) {
    float l = 0.0f;
    #pragma unroll
    for (int w = 0; w < 8; ++w) l += wred[w];
    bcast[0] = m; bcast[1] = l;
  }

  // ---- acc[d] = sum_r p_r * V[r][d]; lane owns d = lane*4..+3 ----
  v4f acc = {};
  #pragma unroll 4
  for (int i = 0; i < CHUNK / 8; ++i) {
    const int r = wave * (CHUNK / 8) + i;
    float p = logits[r];         // LDS broadcast
    v4f v4 = __builtin_nontemporal_load((const v4f*)(V + (size_t)r * HEADDIM + lane * 4));
    acc.x += p * v4.x; acc.y += p * v4.y; acc.z += p * v4.z; acc.w += p * v4.w;
  }
  atomicAdd(&accbuf[lane * 4 + 0], acc.x);   // ds_add_f32
  atomicAdd(&accbuf[lane * 4 + 1], acc.y);
  atomicAdd(&accbuf[lane * 4 + 2], acc.z);
  atomicAdd(&accbuf[lane * 4 + 3], acc.w);
  __syncthreads();

  float* accOut = acc_part + ((size_t)bh * NCHUNK + chunk) * HEADDIM;
  if (threadIdx.x < HEADDIM) accOut[threadIdx.x] = accbuf[threadIdx.x];
  if (threadIdx.x == 0) {
    ml_part[((size_t)bh * NCHUNK + chunk) * 2 + 0] = bcast[0];
    ml_part[((size_t)bh * NCHUNK + chunk) * 2 + 1] = bcast[1];
  }
}

// ---------------------------------------------------------------------------
// Combine: one block (128 thr) per (b,h). Merges 8 chunk partials plus the new
// token treated as partial #9 with m = q.k_new*scale, l = 1, acc = v_new.
// att[b][h*128+d] = (sum_i e^{m_i-M} acc_i[d]) / (sum_i e^{m_i-M} l_i)
// ---------------------------------------------------------------------------
__global__ void attn_combine(const float* __restrict__ q,
                             const float* __restrict__ k_new,
                             const float* __restrict__ v_new,
                             const float* __restrict__ acc_part,
                             const float* __restrict__ ml_part,
                             float* __restrict__ att) {
  const int bh = blockIdx.x;
  const int b = bh >> 5, h = bh & 31;
  const int d = threadIdx.x;
  const size_t qi = (size_t)b * DIM + h * HEADDIM + d;

  __shared__ float red[HEADDIM];
  red[d] = q[qi] * k_new[qi];
  __syncthreads();
  #pragma unroll
  for (int s = 64; s > 0; s >>= 1) {
    if (d < s) red[d] += red[d + s];
    __syncthreads();
  }
  const float logit_new = red[0] * SCALE;

  const float* ml = ml_part + (size_t)bh * NCHUNK * 2;
  float M = logit_new;
  #pragma unroll
  for (int i = 0; i < NCHUNK; ++i) M = fmaxf(M, ml[i * 2]);

  float wn = __expf(logit_new - M);
  float L = wn;                      // l_new = 1
  float o = wn * v_new[qi];          // acc_new = v_new
  #pragma unroll
  for (int i = 0; i < NCHUNK; ++i) {
    float w = __expf(ml[i * 2] - M);
    L += w * ml[i * 2 + 1];
    o += w * acc_part[((size_t)bh * NCHUNK + i) * HEADDIM + d];
  }
  att[qi] = o / L;
}

// ---------------------------------------------------------------------------
// Workspace layout (floats):
//   q[65536] | k_new[65536] | v_new[65536] | att[65536] |
//   acc_part[512*8*128 = 524288] | ml_part[512*8*2 = 8192]   ~3.0 MB total
// ---------------------------------------------------------------------------
extern "C" void kernel_launch(void* const* d_in, const int* in_sizes, int n_in,
                              void* d_out, int out_size, void* d_ws, size_t ws_size,
                              hipStream_t stream) {
  const float* x  = (const float*)d_in[0];
  const float* kc = (const float*)d_in[1];
  const float* vc = (const float*)d_in[2];
  const float* wq = (const float*)d_in[3];
  const float* wk = (const float*)d_in[4];
  const float* wv = (const float*)d_in[5];
  const float* wo = (const float*)d_in[6];
  float* out = (float*)d_out;

  float* ws   = (float*)d_ws;
  float* qb   = ws;
  float* knb  = ws + 65536;
  float* vnb  = ws + 131072;
  float* attb = ws + 196608;
  float* accp = ws + 262144;
  float* mlp  = ws + 786432;

  // zero all atomic-add targets in one launch
  zero_targets<<<(196608 + 65536 + 255) / 256, 256, 0, stream>>>(ws, out);

  // fused QKV projection (WMMA, K-split atomics): 6144 waves in flight
  qkv_gemm_fused<<<768, 256, 0, stream>>>(x, wq, wk, wv, qb, knb, vnb);

  // flash-decode over the 2 GiB KV cache
  attn_partial<<<BATCH * NHEADS * NCHUNK, 256, 0, stream>>>(qb, kc, vc, accp, mlp);
  attn_combine<<<BATCH * NHEADS, HEADDIM, 0, stream>>>(qb, knb, vnb, accp, mlp, attb);

  // output projection (WMMA): 2048 waves
  gemm16_wmma<<<256, 256, 0, stream>>>(attb, wo, out);
}